// QuantumCircuit_19610820674191
// MI455X (gfx1250) — compile-verified
//
#include <hip/hip_runtime.h>
#include <math.h>

// ---------------------------------------------------------------------------
// Quantum state = tensor product of 23 per-qubit 2-vectors:
//   v_q = (1 + cos(th/2),  -i*sin(th/2))       (gate applied to (1,0))
// amp(idx) = [prod of real magnitudes] * (-i)^popcount(idx)
// Split idx = (h << 11) | m  ->  amp = H[h] * L[m]  (rank-1 complex outer
// product, 4096 x 2048).  Computed with V_WMMA_F32_16X16X4_F32 tiles.
//
// Roofline: output 2^23 complex64 = 64 MiB, pure streaming store ->
// ~2.9 us floor at 23.3 TB/s. Compute (64K WMMA ops, ~134 MFLOP) is noise.
// ---------------------------------------------------------------------------

#define NQ     23
#define NLOW   11                 // qubits 0..10  -> L, 2048 entries
#define NHIGH  12                 // qubits 11..22 -> H, 4096 entries
#define LSIZE  (1 << NLOW)        // 2048
#define HSIZE  (1 << NHIGH)       // 4096

typedef __attribute__((ext_vector_type(2))) float v2f;
typedef __attribute__((ext_vector_type(8))) float v8f;

// ---------------------------------------------------------------------------
// Kernel 1: build the two complex factor vectors in workspace.
// ws layout (floats): Lre[2048] | Lim[2048] | Hre[4096] | Him[4096]  = 48 KB
// ---------------------------------------------------------------------------
__global__ void qc_build_factors(const float* __restrict__ params,
                                 float* __restrict__ ws) {
    float* Lre = ws;
    float* Lim = ws + LSIZE;
    float* Hre = ws + 2 * LSIZE;
    float* Him = ws + 2 * LSIZE + HSIZE;

    int tid = blockIdx.x * blockDim.x + threadIdx.x;
    if (tid >= LSIZE + HSIZE) return;

    // (-i)^k phase table, k = popcount mod 4
    const float tre[4] = {1.f, 0.f, -1.f, 0.f};
    const float tim[4] = {0.f, -1.f, 0.f, 1.f};

    if (tid < LSIZE) {
        const int m = tid;
        float prod = 1.f;
#pragma unroll
        for (int q = 0; q < NLOW; ++q) {
            const float half = params[q] * 0.5f;
            const float c = cosf(half);
            const float s = sinf(half);
            prod *= ((m >> q) & 1) ? s : (1.f + c);
        }
        const int k = __popc(m) & 3;
        Lre[m] = prod * tre[k];
        Lim[m] = prod * tim[k];
    } else {
        const int h = tid - LSIZE;
        float prod = 1.f;
#pragma unroll
        for (int b = 0; b < NHIGH; ++b) {
            const float half = params[NLOW + b] * 0.5f;
            const float c = cosf(half);
            const float s = sinf(half);
            prod *= ((h >> b) & 1) ? s : (1.f + c);
        }
        const int k = __popc(h) & 3;
        Hre[h] = prod * tre[k];
        Him[h] = prod * tim[k];
    }
}

// ---------------------------------------------------------------------------
// Kernel 2: rank-1 complex outer product via WMMA f32 16x16x4.
//
// A (16x4, f32): lanes 0-15 hold row M=lane with VGPR0=K0, VGPR1=K1;
//                lanes 16-31 hold K2/K3 -> zeroed via cndmask.
//   K0 = Hre[h0+M], K1 = Him[h0+M]
// B (4x16, f32): lanes 0-15 hold col N=lane with VGPR0=K0, VGPR1=K1.
//   B_re: (Lre, -Lim)  -> D_re = Hre*Lre - Him*Lim
//   B_im: (Lim,  Lre)  -> D_im = Hre*Lim + Him*Lre
//
// Loads are unconditional (addresses valid for all lanes); the lane mask is
// applied as a value-select so no EXEC branching appears in the hot loop and
// EXEC is all-1s at every WMMA (ISA 7.12 requirement).
//
// One wave = one 16-row H tile x eight 16-col L tiles = 2048 complex outputs.
// 4096 waves total (512 blocks x 8 waves) cover all 2^23 amplitudes.
// ---------------------------------------------------------------------------
__global__ void __launch_bounds__(256)
qc_outer_wmma(const float* __restrict__ ws, float* __restrict__ out) {
    const float* Lre = ws;
    const float* Lim = ws + LSIZE;
    const float* Hre = ws + 2 * LSIZE;
    const float* Him = ws + 2 * LSIZE + HSIZE;

    const int lane   = threadIdx.x & 31;
    const int wave   = threadIdx.x >> 5;
    const int task   = blockIdx.x * 8 + wave;   // 0..4095
    const int hTile  = task >> 4;               // 0..255
    const int mGroup = task & 15;               // 0..15 (8 m-tiles each)

    const int  h0  = hTile * 16;
    const bool lo  = lane < 16;
    const int  l15 = lane & 15;

    // A operand: H tile (re, im) in K0/K1 for lanes 0-15, zeros in K2/K3.
    // Unconditional loads (valid for all lanes) + value select -> v_cndmask.
    const float hreAll = Hre[h0 + l15];
    const float himAll = Him[h0 + l15];
    v2f a;
    a.x = lo ? hreAll : 0.f;
    a.y = lo ? himAll : 0.f;

    v2f* o = (v2f*)out;   // interleaved complex64 as 2-float vectors

    for (int t = 0; t < 8; ++t) {
        const int m0 = (mGroup * 8 + t) * 16;

        const float lreAll = Lre[m0 + l15];
        const float limAll = Lim[m0 + l15];
        const float lre = lo ? lreAll : 0.f;
        const float lim = lo ? limAll : 0.f;

        v2f bre, bim;
        bre.x = lre;  bre.y = -lim;   // real part combination
        bim.x = lim;  bim.y =  lre;   // imag part combination

        v8f cz = {};
        // (neg_a, A, neg_b, B, c_mod, C, reuse_a, reuse_b)
        v8f dre = __builtin_amdgcn_wmma_f32_16x16x4_f32(
            false, a, false, bre, (short)0, cz, false, false);
        v8f dim = __builtin_amdgcn_wmma_f32_16x16x4_f32(
            false, a, false, bim, (short)0, cz, false, false);

        // D layout (32-bit 16x16): VGPR v, lanes 0-15 -> (M=v,   N=lane),
        //                                  lanes16-31 -> (M=v+8, N=lane-16)
        const int N     = l15;
        const int Mbase = lo ? 0 : 8;
#pragma unroll
        for (int v = 0; v < 8; ++v) {
            const int    M   = Mbase + v;
            const size_t idx = (size_t)(h0 + M) * LSIZE + (size_t)(m0 + N);
            v2f val;
            val.x = dre[v];
            val.y = dim[v];
            // streaming 64 MiB write, never re-read -> non-temporal store
            __builtin_nontemporal_store(val, &o[idx]);
        }
    }
}

// ---------------------------------------------------------------------------
extern "C" void kernel_launch(void* const* d_in, const int* in_sizes, int n_in,
                              void* d_out, int out_size, void* d_ws, size_t ws_size,
                              hipStream_t stream) {
    const float* params = (const float*)d_in[0];
    float*       ws     = (float*)d_ws;
    float*       out    = (float*)d_out;

    // 1) build L (2048) and H (4096) complex factor vectors: 6144 threads
    qc_build_factors<<<(LSIZE + HSIZE + 255) / 256, 256, 0, stream>>>(params, ws);

    // 2) rank-1 complex outer product -> 2^23 complex64 amplitudes
    //    512 blocks x 256 threads (8 waves/block, 1 wave-task each)
    qc_outer_wmma<<<512, 256, 0, stream>>>(ws, out);
}